// DCell_89206470738571
// MI455X (gfx1250) — compile-verified
//
#include <hip/hip_runtime.h>
#include <hip/hip_bf16.h>

typedef __attribute__((ext_vector_type(2))) float v2f;
typedef __attribute__((ext_vector_type(8))) float v8f;
typedef __attribute__((ext_vector_type(4))) unsigned int uint32x4;
typedef __attribute__((ext_vector_type(4))) int int32x4;
typedef __attribute__((ext_vector_type(8))) int int32x8;

// Problem sizes
constexpr int S_  = 512;   // leaf subsystems
constexpr int B_  = 2048;  // batch
constexpr int GL_ = 32;    // leaf genes
constexpr int OL_ = 20;    // leaf out
constexpr int M_  = 32;    // mid subsystems
constexpr int C_  = 16;    // children per mid
constexpr int GM_ = 64;    // mid genes
constexpr int OM_ = 20;    // mid out
constexpr int GR_ = 128;   // root genes
constexpr int OR2 = 38;    // root out
constexpr int IM_ = C_ * OL_ + GM_; // 384
constexpr int IR_ = M_ * OM_ + GR_; // 768
constexpr float EPSI = 1e-5f;
constexpr float INVB = 1.0f / (float)B_;

// padded N for packed WMMA B-layout
constexpr int NPL = 32;  // leaf/mid (OL=OM=20 -> 32)
constexpr int NPR = 48;  // root (OR=38 -> 48)

__device__ __forceinline__ v8f wmma4(v2f a, v2f b, v8f c) {
  return __builtin_amdgcn_wmma_f32_16x16x4_f32(false, a, false, b, (short)0, c,
                                               false, false);
}

// Branchless tanh: clamp, hardware exp, hardware rcp. abs err ~1e-7.
__device__ __forceinline__ float fast_tanh(float x) {
  float xc = fminf(fmaxf(x, -12.f), 12.f);
  float e  = __expf(2.f * xc);
  return (e - 1.f) * __builtin_amdgcn_rcpf(e + 1.f);
}

// ---------------------------------------------------------------------------
// Pack W[sub][K][N] -> Wt[sub][K/2][NP][2] (WMMA B operand pairs, zero padded)
// ---------------------------------------------------------------------------
__global__ void pack_w(const float* __restrict__ W, float* __restrict__ Wt,
                       int nsub, int K, int N, int NP) {
  int i = blockIdx.x * blockDim.x + threadIdx.x;
  int total = nsub * (K / 2) * NP;
  if (i >= total) return;
  int n   = i % NP;
  int t   = (i / NP) % (K / 2);
  int sub = i / (NP * (K / 2));
  float2 v;
  if (n < N) {
    v.x = W[((size_t)sub * K + 2 * t) * N + n];
    v.y = W[((size_t)sub * K + 2 * t + 1) * N + n];
  } else {
    v.x = 0.f; v.y = 0.f;
  }
  ((float2*)Wt)[i] = v;
}

// ---------------------------------------------------------------------------
// Stage 1: leaf GEMM. TDM stages a 128x32 f32 x-tile into LDS (padded rows),
// 8 waves consume it. One block per (s, 128-row slab): 512*16 = 8192 blocks.
// ---------------------------------------------------------------------------
__global__ __launch_bounds__(256) void leaf_gemm(
    const float* __restrict__ x_leaf, const float* __restrict__ wt_leaf,
    const float* __restrict__ b_leaf, float* __restrict__ hbuf,
    float* __restrict__ lsum, float* __restrict__ lsq) {
  __shared__ float tileA[128 * 34];  // 34-float row stride via TDM pad

  const int lane = threadIdx.x & 31;
  const int wid  = threadIdx.x >> 5;
  const int s    = blockIdx.x >> 4;
  const int row0 = (blockIdx.x & 15) << 7;  // 128-row slab
  const int mrow = lane & 15;
  const int hk   = lane >> 4;
  const int nA   = lane & 15;

  if (wid == 0) {
    unsigned long long ga =
        (unsigned long long)(const void*)(x_leaf + ((size_t)s * B_ + row0) * GL_);
    unsigned int ldsoff = (unsigned int)(uintptr_t)&tileA[0];
    uint32x4 g0;
    g0.x = 1u;                              // count=1 (user descriptor)
    g0.y = ldsoff;                          // lds_addr
    g0.z = (unsigned int)ga;                // global_addr[31:0]
    g0.w = (unsigned int)((ga >> 32) & 0x1FFFFFFu) | 0x80000000u;  // type=2
    // data_size=4B, pad_enable, pad_interval=32dw, pad_amount=2dw,
    // tensor_dim0=32, tensor_dim1=128, tile_dim0=32, tile_dim1=128, stride0=32
    int32x8 g1 = {0x03120000, 0x00200000, 0x00800000, 0x00200000, 128, 32, 0, 0};
    int32x4 gz = {0, 0, 0, 0};
#if __clang_major__ >= 23
    int32x8 gz8 = {0, 0, 0, 0, 0, 0, 0, 0};
    __builtin_amdgcn_tensor_load_to_lds(g0, g1, gz, gz, gz8, 0);
#else
    __builtin_amdgcn_tensor_load_to_lds(g0, g1, gz, gz, 0);
#endif
    __builtin_amdgcn_s_wait_tensorcnt(0);
  }
  __syncthreads();

  const float* Wt = wt_leaf + (size_t)s * (GL_ / 2) * NPL * 2;
  const float* As = &tileA[(wid * 16 + mrow) * 34];

  v8f c0 = {}; v8f c1 = {};
#pragma unroll
  for (int k = 0; k < GL_; k += 4) {
    const int ka = k + 2 * hk;
    v2f a = *(const v2f*)(As + ka);
    const float* wp = Wt + (size_t)(k / 2 + hk) * (NPL * 2);
    v2f b0 = *(const v2f*)(wp + nA * 2);
    v2f b1 = *(const v2f*)(wp + (16 + nA) * 2);
    c0 = wmma4(a, b0, c0);
    c1 = wmma4(a, b1, c1);
  }

  // epilogue: bias + tanh, store h, per-column batch partial sums
  const int rbase = row0 + wid * 16 + 8 * hk;
  float* H = hbuf + ((size_t)s * B_ + rbase) * OL_;
  const float bias0 = b_leaf[s * OL_ + nA];
  float s0 = 0.f, q0 = 0.f, s1 = 0.f, q1 = 0.f;
#pragma unroll
  for (int r = 0; r < 8; ++r) {
    float h = fast_tanh(c0[r] + bias0);
    H[r * OL_ + nA] = h;
    s0 += h; q0 += h * h;
  }
  if (nA < OL_ - 16) {
    const float bias1 = b_leaf[s * OL_ + 16 + nA];
#pragma unroll
    for (int r = 0; r < 8; ++r) {
      float h = fast_tanh(c1[r] + bias1);
      H[r * OL_ + 16 + nA] = h;
      s1 += h; q1 += h * h;
    }
  }
  s0 += __shfl_xor(s0, 16, 32); q0 += __shfl_xor(q0, 16, 32);
  s1 += __shfl_xor(s1, 16, 32); q1 += __shfl_xor(q1, 16, 32);
  if (lane < 16) {
    atomicAdd(&lsum[s * OL_ + nA], s0);
    atomicAdd(&lsq [s * OL_ + nA], q0);
    if (nA < OL_ - 16) {
      atomicAdd(&lsum[s * OL_ + 16 + nA], s1);
      atomicAdd(&lsq [s * OL_ + 16 + nA], q1);
    }
  }
}

// ---------------------------------------------------------------------------
// BN finalize: alpha = g*rsqrt(var+eps), beta = be - alpha*mean
// ---------------------------------------------------------------------------
__global__ void bn_finalize(const float* __restrict__ sum, const float* __restrict__ sq,
                            const float* __restrict__ g, const float* __restrict__ be,
                            float* __restrict__ alpha, float* __restrict__ beta, int n) {
  int i = blockIdx.x * blockDim.x + threadIdx.x;
  if (i >= n) return;
  float mean = sum[i] * INVB;
  float var  = sq[i] * INVB - mean * mean;
  float a    = g[i] * rsqrtf(var + EPSI);
  alpha[i] = a;
  beta[i]  = be[i] - a * mean;
}

// ---------------------------------------------------------------------------
// Stage 2: mid GEMM. A gathered from [x_mid | BN(leaf h)] (uniform branches).
// one wave per (m, 16-row tile): 4096 waves
// ---------------------------------------------------------------------------
__global__ void mid_gemm(const float* __restrict__ x_mid,
                         const float* __restrict__ hbuf,
                         const float* __restrict__ lal, const float* __restrict__ lbe,
                         const float* __restrict__ wt_mid,
                         const float* __restrict__ b_mid,
                         float* __restrict__ hmid,
                         float* __restrict__ msum, float* __restrict__ msq) {
  const int lane = threadIdx.x & 31;
  const int gw   = blockIdx.x * (blockDim.x >> 5) + (threadIdx.x >> 5);
  const int m    = gw >> 7;
  const int row0 = (gw & 127) << 4;
  const int mrow = lane & 15;
  const int hk   = lane >> 4;
  const int nA   = lane & 15;
  const int row  = row0 + mrow;

  const float* XM = x_mid + (size_t)m * B_ * GM_;
  const float* WT = wt_mid + (size_t)m * (IM_ / 2) * NPL * 2;

  v8f c0 = {}; v8f c1 = {};
  for (int k = 0; k < IM_; k += 4) {
    const int ka = k + 2 * hk;
    v2f a;
    if (ka < GM_) {
      a = *(const v2f*)(XM + (size_t)row * GM_ + ka);
    } else {
      int j  = ka - GM_;
      int cc = j / OL_;
      int o  = j % OL_;                   // j even -> o even -> pair stays in child
      int sl = m * C_ + cc;
      v2f hv = *(const v2f*)(hbuf + ((size_t)sl * B_ + row) * OL_ + o);
      v2f al = *(const v2f*)(lal + sl * OL_ + o);
      v2f bt = *(const v2f*)(lbe + sl * OL_ + o);
      a.x = al.x * hv.x + bt.x;
      a.y = al.y * hv.y + bt.y;
    }
    const float* wp = WT + (size_t)(k / 2 + hk) * (NPL * 2);
    v2f b0 = *(const v2f*)(wp + nA * 2);
    v2f b1 = *(const v2f*)(wp + (16 + nA) * 2);
    c0 = wmma4(a, b0, c0);
    c1 = wmma4(a, b1, c1);
  }

  float* H = hmid + ((size_t)m * B_ + row0 + 8 * hk) * OM_;
  const float bias0 = b_mid[m * OM_ + nA];
  float s0 = 0.f, q0 = 0.f, s1 = 0.f, q1 = 0.f;
#pragma unroll
  for (int r = 0; r < 8; ++r) {
    float h = fast_tanh(c0[r] + bias0);
    H[r * OM_ + nA] = h;
    s0 += h; q0 += h * h;
  }
  if (nA < OM_ - 16) {
    const float bias1 = b_mid[m * OM_ + 16 + nA];
#pragma unroll
    for (int r = 0; r < 8; ++r) {
      float h = fast_tanh(c1[r] + bias1);
      H[r * OM_ + 16 + nA] = h;
      s1 += h; q1 += h * h;
    }
  }
  s0 += __shfl_xor(s0, 16, 32); q0 += __shfl_xor(q0, 16, 32);
  s1 += __shfl_xor(s1, 16, 32); q1 += __shfl_xor(q1, 16, 32);
  if (lane < 16) {
    atomicAdd(&msum[m * OM_ + nA], s0);
    atomicAdd(&msq [m * OM_ + nA], q0);
    if (nA < OM_ - 16) {
      atomicAdd(&msum[m * OM_ + 16 + nA], s1);
      atomicAdd(&msq [m * OM_ + 16 + nA], q1);
    }
  }
}

// ---------------------------------------------------------------------------
// Stage 3: root GEMM. A gathered from [x_root | BN(mid h)], 3 N-tiles.
// one wave per 16-row tile: 128 waves
// ---------------------------------------------------------------------------
__global__ void root_gemm(const float* __restrict__ x_root,
                          const float* __restrict__ hmid,
                          const float* __restrict__ mal, const float* __restrict__ mbe,
                          const float* __restrict__ wt_root,
                          const float* __restrict__ b_root,
                          float* __restrict__ hroot,
                          float* __restrict__ rsum, float* __restrict__ rsq) {
  const int lane = threadIdx.x & 31;
  const int gw   = blockIdx.x * (blockDim.x >> 5) + (threadIdx.x >> 5);
  const int row0 = gw << 4;
  const int mrow = lane & 15;
  const int hk   = lane >> 4;
  const int nA   = lane & 15;
  const int row  = row0 + mrow;

  v8f c0 = {}; v8f c1 = {}; v8f c2 = {};
  for (int k = 0; k < IR_; k += 4) {
    const int ka = k + 2 * hk;
    v2f a;
    if (ka < GR_) {
      a = *(const v2f*)(x_root + (size_t)row * GR_ + ka);
    } else {
      int j  = ka - GR_;
      int mm = j / OM_;
      int o  = j % OM_;                   // j even -> o even
      v2f hv = *(const v2f*)(hmid + ((size_t)mm * B_ + row) * OM_ + o);
      v2f al = *(const v2f*)(mal + mm * OM_ + o);
      v2f bt = *(const v2f*)(mbe + mm * OM_ + o);
      a.x = al.x * hv.x + bt.x;
      a.y = al.y * hv.y + bt.y;
    }
    const float* wp = wt_root + (size_t)(k / 2 + hk) * (NPR * 2);
    v2f b0 = *(const v2f*)(wp + nA * 2);
    v2f b1 = *(const v2f*)(wp + (16 + nA) * 2);
    v2f b2 = *(const v2f*)(wp + (32 + nA) * 2);
    c0 = wmma4(a, b0, c0);
    c1 = wmma4(a, b1, c1);
    c2 = wmma4(a, b2, c2);
  }

  float* H = hroot + (size_t)(row0 + 8 * hk) * OR2;
  const float bias0 = b_root[nA];
  const float bias1 = b_root[16 + nA];
  float s0 = 0.f, q0 = 0.f, s1 = 0.f, q1 = 0.f, s2 = 0.f, q2 = 0.f;
#pragma unroll
  for (int r = 0; r < 8; ++r) {
    float h0 = fast_tanh(c0[r] + bias0);
    float h1 = fast_tanh(c1[r] + bias1);
    H[r * OR2 + nA]      = h0;
    H[r * OR2 + 16 + nA] = h1;
    s0 += h0; q0 += h0 * h0;
    s1 += h1; q1 += h1 * h1;
  }
  if (nA < OR2 - 32) {
    const float bias2 = b_root[32 + nA];
#pragma unroll
    for (int r = 0; r < 8; ++r) {
      float h = fast_tanh(c2[r] + bias2);
      H[r * OR2 + 32 + nA] = h;
      s2 += h; q2 += h * h;
    }
  }
  s0 += __shfl_xor(s0, 16, 32); q0 += __shfl_xor(q0, 16, 32);
  s1 += __shfl_xor(s1, 16, 32); q1 += __shfl_xor(q1, 16, 32);
  s2 += __shfl_xor(s2, 16, 32); q2 += __shfl_xor(q2, 16, 32);
  if (lane < 16) {
    atomicAdd(&rsum[nA], s0);      atomicAdd(&rsq[nA], q0);
    atomicAdd(&rsum[16 + nA], s1); atomicAdd(&rsq[16 + nA], q1);
    if (nA < OR2 - 32) { atomicAdd(&rsum[32 + nA], s2); atomicAdd(&rsq[32 + nA], q2); }
  }
}

// ---------------------------------------------------------------------------
__global__ void zero_kernel(float* __restrict__ p, int n) {
  int i = blockIdx.x * blockDim.x + threadIdx.x;
  if (i < n) p[i] = 0.f;
}

__global__ void root_out(const float* __restrict__ hroot,
                         const float* __restrict__ ral, const float* __restrict__ rbe,
                         float* __restrict__ out) {
  int i = blockIdx.x * blockDim.x + threadIdx.x;
  if (i < B_ * OR2) {
    int o = i % OR2;
    out[i] = ral[o] * hroot[i] + rbe[o];
  }
}

// ---------------------------------------------------------------------------
extern "C" void kernel_launch(void* const* d_in, const int* in_sizes, int n_in,
                              void* d_out, int out_size, void* d_ws, size_t ws_size,
                              hipStream_t stream) {
  const float* x_leaf  = (const float*)d_in[0];
  const float* x_mid   = (const float*)d_in[1];
  const float* x_root  = (const float*)d_in[2];
  const float* W_leaf  = (const float*)d_in[3];
  const float* b_leaf  = (const float*)d_in[4];
  const float* g_leaf  = (const float*)d_in[5];
  const float* be_leaf = (const float*)d_in[6];
  const float* W_mid   = (const float*)d_in[7];
  const float* b_mid   = (const float*)d_in[8];
  const float* g_mid   = (const float*)d_in[9];
  const float* be_mid  = (const float*)d_in[10];
  const float* W_root  = (const float*)d_in[11];
  const float* b_root  = (const float*)d_in[12];
  const float* g_root  = (const float*)d_in[13];
  const float* be_root = (const float*)d_in[14];
  float* out = (float*)d_out;

  // Workspace layout (floats)
  float* ws    = (float*)d_ws;
  float* hbuf  = ws;                                   // [S,B,OL]
  float* hmid  = hbuf  + (size_t)S_ * B_ * OL_;        // [M,B,OM]
  float* hroot = hmid  + (size_t)M_ * B_ * OM_;        // [B,OR]
  float* stats = hroot + (size_t)B_ * OR2;
  float* lsum = stats;               float* lsq = lsum + S_ * OL_;
  float* lal  = lsq  + S_ * OL_;     float* lbe = lal  + S_ * OL_;
  float* msum = lbe  + S_ * OL_;     float* msq = msum + M_ * OM_;
  float* mal  = msq  + M_ * OM_;     float* mbe = mal  + M_ * OM_;
  float* rsum = mbe  + M_ * OM_;     float* rsq = rsum + 64;
  float* ral  = rsq  + 64;           float* rbe = ral  + 64;
  const int nstats = 4 * S_ * OL_ + 4 * M_ * OM_ + 4 * 64; // 43776
  float* wtL = rbe + 64;                               // [S][16][32][2]
  float* wtM = wtL + (size_t)S_ * (GL_ / 2) * NPL * 2; // [M][192][32][2]
  float* wtR = wtM + (size_t)M_ * (IM_ / 2) * NPL * 2; // [384][48][2]

  zero_kernel<<<(nstats + 255) / 256, 256, 0, stream>>>(stats, nstats);

  const int nwL = S_ * (GL_ / 2) * NPL;
  const int nwM = M_ * (IM_ / 2) * NPL;
  const int nwR = (IR_ / 2) * NPR;
  pack_w<<<(nwL + 255) / 256, 256, 0, stream>>>(W_leaf, wtL, S_, GL_, OL_, NPL);
  pack_w<<<(nwM + 255) / 256, 256, 0, stream>>>(W_mid,  wtM, M_, IM_, OM_, NPL);
  pack_w<<<(nwR + 255) / 256, 256, 0, stream>>>(W_root, wtR, 1,  IR_, OR2, NPR);

  // leaf: 8192 blocks x 8 waves, TDM-staged A tiles
  leaf_gemm<<<8192, 256, 0, stream>>>(x_leaf, wtL, b_leaf, hbuf, lsum, lsq);
  bn_finalize<<<(S_ * OL_ + 255) / 256, 256, 0, stream>>>(lsum, lsq, g_leaf, be_leaf,
                                                          lal, lbe, S_ * OL_);
  // mid: 4096 waves
  mid_gemm<<<512, 256, 0, stream>>>(x_mid, hbuf, lal, lbe, wtM, b_mid, hmid, msum, msq);
  bn_finalize<<<(M_ * OM_ + 255) / 256, 256, 0, stream>>>(msum, msq, g_mid, be_mid,
                                                          mal, mbe, M_ * OM_);
  // root: 128 waves
  root_gemm<<<16, 256, 0, stream>>>(x_root, hmid, mal, mbe, wtR, b_root, hroot, rsum, rsq);
  bn_finalize<<<1, 64, 0, stream>>>(rsum, rsq, g_root, be_root, ral, rbe, OR2);

  root_out<<<(B_ * OR2 + 255) / 256, 256, 0, stream>>>(hroot, ral, rbe, out);
}